// VoxelFieldFusion_48223892799593
// MI455X (gfx1250) — compile-verified
//
#include <hip/hip_runtime.h>
#include <hip/hip_bf16.h>

// ---------------------------------------------------------------------------
// VoxelFieldFusion for MI455X (gfx1250, wave32)
//   Memory-bound op: 360MB grid output dominates (~31us at 23.3TB/s roofline).
//   The two dense GEMMs (~1.6 GFLOP total) run on the WMMA pipe in fp32
//   (V_WMMA_F32_16X16X4_F32) with WEIGHTS as the A operand held resident in
//   VGPRs, activations as B (coalesced/LDS loads, prefetched into register
//   arrays so WMMAs issue back-to-back instead of load->wait->wmma chains).
// ---------------------------------------------------------------------------

#define DZc 10
#define DYc 400
#define DXc 352
#define Cc  64

typedef __attribute__((ext_vector_type(2))) float v2f;
typedef __attribute__((ext_vector_type(8))) float v8f;

// D(16x16,f32) = A(16x4,f32) x B(4x16,f32) + C
// Fragment layout (ISA 7.12.2):
//   A: lane<16 -> {A[l][0],A[l][1]}, lane>=16 -> {A[l-16][2],A[l-16][3]}
//   B: vgpr0: lane<16 B[0][n], lane>=16 B[2][n]; vgpr1: B[1][n], B[3][n]
//   C/D: vgpr i: lane<16 -> (M=i,N=lane), lane>=16 -> (M=i+8,N=lane-16)
__device__ __forceinline__ v8f wmma_f32_4(v2f a, v2f b, v8f c) {
  return __builtin_amdgcn_wmma_f32_16x16x4_f32(
      /*neg_a=*/false, a, /*neg_b=*/false, b,
      /*c_mod=*/(short)0, c, /*reuse_a=*/false, /*reuse_b=*/false);
}

// ---------------------------------------------------------------------------
// Kernel 1: zero the 360MB output grid (b128 stores, grid-stride)
// ---------------------------------------------------------------------------
__global__ void vff_zero_kernel(float4* __restrict__ out, long n4) {
  const float4 z = make_float4(0.f, 0.f, 0.f, 0.f);
  const long stride = (long)gridDim.x * blockDim.x;
  for (long i = (long)blockIdx.x * blockDim.x + threadIdx.x; i < n4; i += stride)
    out[i] = z;
}

// ---------------------------------------------------------------------------
// Kernel 2: img_proc_t[p][o] = sum_c img_w[o][c]*image_feat[c][p] + img_b[o]
//   WMMA with D = (M=out-channel) x (N=pixel):
//     A = img_w tile (16o x 4c)      -> resident in VGPRs for whole kernel
//     B = image_feat (4c x 16p)      -> coalesced b32 loads, prefetched
//     D store: lane owns 8 consecutive channels of one pixel -> 2x b128.
//   Output is PIXEL-major so the per-ray channel gather later is one 256B row.
// ---------------------------------------------------------------------------
__global__ void vff_imgproc_kernel(const float* __restrict__ image_feat,
                                   const float* __restrict__ img_w,
                                   const float* __restrict__ img_b,
                                   float* __restrict__ img_proc_t,
                                   int HW) {
  const int lane = threadIdx.x & 31;
  const int wave = threadIdx.x >> 5;   // 0..3 -> output channel tile
  const int o0   = wave * 16;
  const int half = lane >> 4;
  const int ln   = lane & 15;

  // Resident A fragments (weights): A[o][c] = img_w[o*64 + c]
  v2f afrag[16];
#pragma unroll
  for (int s = 0; s < 16; ++s) {
    const float* wr = img_w + (o0 + ln) * Cc + 4 * s + 2 * half;
    afrag[s].x = wr[0];
    afrag[s].y = wr[1];
  }
  // Bias for the 8 output channels this lane holds (rows o0+8*half .. +7)
  v8f biasv;
#pragma unroll
  for (int r = 0; r < 8; ++r) biasv[r] = img_b[o0 + 8 * half + r];

  const int ntile = HW >> 4;
  for (int t = blockIdx.x; t < ntile; t += gridDim.x) {
    const int p = t * 16 + ln;         // this lane's pixel column

    // Prefetch all B fragments: coalesced, 32 loads in flight
    v2f bfrag[16];
#pragma unroll
    for (int s = 0; s < 16; ++s) {
      const int c0 = 4 * s + 2 * half;
      bfrag[s].x = image_feat[(long)c0 * HW + p];
      bfrag[s].y = image_feat[(long)(c0 + 1) * HW + p];
    }

    v8f acc = {};
#pragma unroll
    for (int s = 0; s < 16; ++s) acc = wmma_f32_4(afrag[s], bfrag[s], acc);

    // Store: 8 consecutive channels of pixel p -> two b128 stores
    float* dst = img_proc_t + (long)p * Cc + o0 + 8 * half;
    float4 lo, hi;
    lo.x = acc[0] + biasv[0]; lo.y = acc[1] + biasv[1];
    lo.z = acc[2] + biasv[2]; lo.w = acc[3] + biasv[3];
    hi.x = acc[4] + biasv[4]; hi.y = acc[5] + biasv[5];
    hi.z = acc[6] + biasv[6]; hi.w = acc[7] + biasv[7];
    *(float4*)(dst)     = lo;
    *(float4*)(dst + 4) = hi;
  }
}

// ---------------------------------------------------------------------------
// Kernel 3: voxel path — gather image_feat at image_grid, masked add to
//           voxel_feat, atomic scatter into grid. Pure VMEM, one (n,c)/thread.
//           image_feat (7.9MB) is L2-resident, so the strided gather is cheap.
// ---------------------------------------------------------------------------
__global__ void vff_voxel_kernel(const float* __restrict__ image_feat,
                                 const float* __restrict__ voxel_feat,
                                 const int* __restrict__ image_grid,
                                 const unsigned char* __restrict__ point_mask,
                                 const int* __restrict__ voxel_zyx,
                                 float* __restrict__ grid,
                                 int N, int HW, int W) {
  const long total  = (long)N * Cc;
  const long stride = (long)gridDim.x * blockDim.x;
  for (long i = (long)blockIdx.x * blockDim.x + threadIdx.x; i < total; i += stride) {
    const int n = (int)(i >> 6);
    const int c = (int)(i & 63);
    float v = voxel_feat[i];
    if (point_mask[n]) {
      const int gx = image_grid[n * 2 + 0];
      const int gy = image_grid[n * 2 + 1];
      v += image_feat[(long)c * HW + gy * W + gx];
    }
    const int* z = voxel_zyx + n * 3;
    const long idx = ((long)z[0] * DYc + z[1]) * DXc + z[2];
    atomicAdd(grid + idx * Cc + c, v);
  }
}

// ---------------------------------------------------------------------------
// Kernel 4: ray path. Block = 128 threads (4 waves) per 16-ray tile.
//   Stage fuse_in (16 x 128) in LDS (stride 132 -> conflict-free ds_load_b64).
//   WMMA with D = (M=out-channel) x (N=ray):
//     A = fuse_w tile (16o x 4k)  -> resident in VGPRs (64 VGPRs, 32 steps)
//     B = fuse_in^T from LDS      -> v2f ds_load_b64, prefetched (32 frags)
//   Epilogue: one mask/index load per lane; 8 consecutive-channel atomics.
// ---------------------------------------------------------------------------
#define LDS_STRIDE 132

__global__ void vff_ray_kernel(const float* __restrict__ ray_points,
                               const int* __restrict__ ray_grid,
                               const int* __restrict__ ray_voxel,
                               const unsigned char* __restrict__ ray_mask,
                               const float* __restrict__ ray_w,
                               const float* __restrict__ ray_b,
                               const float* __restrict__ fuse_w,
                               const float* __restrict__ fuse_b,
                               const float* __restrict__ img_proc_t,
                               float* __restrict__ grid,
                               int ntile, int W) {
  __shared__ float lds_in[16 * LDS_STRIDE];

  const int tid  = threadIdx.x;
  const int lane = tid & 31;
  const int wave = tid >> 5;       // 0..3 -> output channel tile
  const int o0   = wave * 16;
  const int half = lane >> 4;
  const int ln   = lane & 15;

  // Resident A fragments (weights): A[o][k] = fuse_w[o*128 + k], 32 K-steps
  v2f afrag[32];
#pragma unroll
  for (int s = 0; s < 32; ++s) {
    const float* wr = fuse_w + (o0 + ln) * (2 * Cc) + 4 * s + 2 * half;
    afrag[s].x = wr[0];
    afrag[s].y = wr[1];
  }
  // Bias for the 8 output channels this lane holds
  v8f biasv;
#pragma unroll
  for (int r = 0; r < 8; ++r) biasv[r] = fuse_b[o0 + 8 * half + r];

  for (int t = blockIdx.x; t < ntile; t += gridDim.x) {
    const int m0 = t * 16;

    // ---- stage fuse_in[16][128] into LDS: thread -> (row m, 16-wide k seg)
    {
      const int m    = tid >> 3;       // 0..15
      const int kseg = tid & 7;        // 8 segments of 16
      const int gm   = m0 + m;
      if (kseg < 4) {                  // ray_feat = ray_points @ ray_w.T + ray_b
        const float p0 = ray_points[gm * 3 + 0];
        const float p1 = ray_points[gm * 3 + 1];
        const float p2 = ray_points[gm * 3 + 2];
#pragma unroll
        for (int j = 0; j < 16; ++j) {
          const int k = kseg * 16 + j;
          lds_in[m * LDS_STRIDE + k] =
              p0 * ray_w[k * 3 + 0] + p1 * ray_w[k * 3 + 1] +
              p2 * ray_w[k * 3 + 2] + ray_b[k];
        }
      } else {                         // img_gather: contiguous 64B from ws
        const int gx = ray_grid[gm * 2 + 0];
        const int gy = ray_grid[gm * 2 + 1];
        const float* gb = img_proc_t + (long)(gy * W + gx) * Cc + (kseg - 4) * 16;
#pragma unroll
        for (int j = 0; j < 16; ++j)
          lds_in[m * LDS_STRIDE + 64 + (kseg - 4) * 16 + j] = gb[j];
      }
    }
    __syncthreads();

    // ---- prefetch B fragments from LDS (b64, conflict-free via stride 132)
    v2f bfrag[32];
#pragma unroll
    for (int s = 0; s < 32; ++s) {
      const int k0 = 4 * s + 2 * half;
      bfrag[s] = *reinterpret_cast<const v2f*>(&lds_in[ln * LDS_STRIDE + k0]);
    }

    // ---- fuse GEMM: 32 back-to-back WMMAs, K = 128
    v8f acc = {};
#pragma unroll
    for (int s = 0; s < 32; ++s) acc = wmma_f32_4(afrag[s], bfrag[s], acc);

    // ---- epilogue: bias, ReLU, mask, atomic scatter
    const int gm = m0 + ln;            // this lane's ray
    if (ray_mask[gm]) {
      const int* rv = ray_voxel + gm * 3;
      const long idx = ((long)rv[0] * DYc + rv[1]) * DXc + rv[2];
      float* dst = grid + idx * Cc + o0 + 8 * half;
#pragma unroll
      for (int r = 0; r < 8; ++r) {
        const float v = fmaxf(acc[r] + biasv[r], 0.f);
        atomicAdd(dst + r, v);
      }
    }
    __syncthreads();
  }
}

// ---------------------------------------------------------------------------
extern "C" void kernel_launch(void* const* d_in, const int* in_sizes, int n_in,
                              void* d_out, int out_size, void* d_ws, size_t ws_size,
                              hipStream_t stream) {
  const float*         image_feat = (const float*)d_in[0];
  const float*         voxel_feat = (const float*)d_in[1];
  /* d_in[2] voxel_batch: all zeros (single batch) -> unused */
  const int*           voxel_zyx  = (const int*)d_in[3];
  const int*           image_grid = (const int*)d_in[4];
  const unsigned char* point_mask = (const unsigned char*)d_in[5];
  const float*         ray_points = (const float*)d_in[6];
  const int*           ray_grid   = (const int*)d_in[7];
  const int*           ray_voxel  = (const int*)d_in[8];
  const unsigned char* ray_mask   = (const unsigned char*)d_in[9];
  const float*         ray_w      = (const float*)d_in[10];
  const float*         ray_b      = (const float*)d_in[11];
  const float*         img_w      = (const float*)d_in[12];
  const float*         img_b      = (const float*)d_in[13];
  const float*         fuse_w     = (const float*)d_in[14];
  const float*         fuse_b     = (const float*)d_in[15];

  float* grid       = (float*)d_out;
  float* img_proc_t = (float*)d_ws;   // [HW][64] = 7.86 MB, pixel-major

  const int HW = in_sizes[0] / Cc;    // 96*320 = 30720
  const int W  = 320;
  const int N  = in_sizes[1] / Cc;    // 80000
  const int M  = in_sizes[6] / 3;     // 80000

  // 1) zero the 360MB grid (harness poisons once; we must re-zero each call)
  const long n4 = (long)out_size / 4;
  vff_zero_kernel<<<4096, 256, 0, stream>>>((float4*)d_out, n4);

  // 2) 1x1 conv on the image (WMMA f32), pixel-major output to workspace
  vff_imgproc_kernel<<<HW / 16, 128, 0, stream>>>(image_feat, img_w, img_b,
                                                  img_proc_t, HW);

  // 3) voxel gather + scatter-add
  vff_voxel_kernel<<<8192, 256, 0, stream>>>(image_feat, voxel_feat, image_grid,
                                             point_mask, voxel_zyx, grid, N, HW, W);

  // 4) ray MLP fusion (WMMA f32) + scatter-add
  vff_ray_kernel<<<M / 16, 128, 0, stream>>>(ray_points, ray_grid, ray_voxel,
                                             ray_mask, ray_w, ray_b, fuse_w,
                                             fuse_b, img_proc_t, grid, M / 16, W);
}